// RMILoss_23673859736483
// MI455X (gfx1250) — compile-verified
//
#include <hip/hip_runtime.h>
#include <hip/hip_bf16.h>
#include <math.h>

// Problem constants (from reference)
#define NUM_CLASSES 26
#define NBATCH 4
#define H 512
#define W 512
#define PH 171              // pooled height: floor((512+2-3)/3)+1
#define PW 171
#define NH 169              // PH - (RADIUS-1)
#define NW 169
#define NC (NBATCH*NUM_CLASSES)   // 104
#define POOL_ELEMS (PH*PW)        // 29241
#define CHUNKS 16
#define ROWS_PER_CHUNK ((NH + CHUNKS - 1)/CHUNKS)  // 11

typedef __attribute__((ext_vector_type(2))) float v2f;
typedef __attribute__((ext_vector_type(8))) float v8f;

// ---------------------------------------------------------------- zero grams
__global__ void RMI_zero_kernel(float* __restrict__ p, int n) {
    int t = blockIdx.x * blockDim.x + threadIdx.x;
    if (t < n) p[t] = 0.f;
}

// ------------------------------------------- fused sigmoid+mask+3x3/3 maxpool
__global__ void RMI_pool_kernel(const float* __restrict__ score,
                                const float* __restrict__ target,
                                const float* __restrict__ mask,
                                float* __restrict__ pla,
                                float* __restrict__ ppr) {
    int t = blockIdx.x * blockDim.x + threadIdx.x;
    if (t >= NC * POOL_ELEMS) return;
    int nc = t / POOL_ELEMS;
    int ij = t - nc * POOL_ELEMS;
    int i  = ij / PW;
    int j  = ij - i * PW;
    int n  = nc / NUM_CLASSES;

    const float* sc = score  + (size_t)nc * (H * W);
    const float* tg = target + (size_t)nc * (H * W);
    const float* mk = mask   + (size_t)n  * (H * W);

    int r0 = 3 * i - 1, c0 = 3 * j - 1;   // padding (1,1), stride 3
    float mla = -1e30f, mpr = -1e30f;
    #pragma unroll
    for (int dr = 0; dr < 3; ++dr) {
        int r = r0 + dr;
        if (r < 0 || r >= H) continue;
        #pragma unroll
        for (int dc = 0; dc < 3; ++dc) {
            int c = c0 + dc;
            if (c < 0 || c >= W) continue;
            int idx = r * W + c;
            float m  = mk[idx];
            float la = tg[idx] * m;
            float p  = (1.f / (1.f + __expf(-sc[idx]))) * m + 1e-6f;
            mla = fmaxf(mla, la);
            mpr = fmaxf(mpr, p);
        }
    }
    pla[t] = mla;
    ppr[t] = mpr;
}

// ------------------------------------- WMMA Gram accumulation (f32 16x16x4)
// Per (n,c): rows 0..8 of X = the 9 shifted views, row 9 = 1.0 (gives the
// per-row sums and the sample count in the Gram for free), rows 10..15 = 0.
// For f32 16x4 A and 4x16 B the per-lane layouts coincide:
//   lane L supplies X[L&15][m0 + 2*(L>>4) + j]  (j = VGPR index 0/1)
// so one v2f serves as both the A and the B operand.
// One wave per block; chunk bounds from blockIdx -> scalar (uniform) loop,
// EXEC provably all-ones at every WMMA, zero divergence in the hot loop.
__global__ __launch_bounds__(32)
void RMI_gram_kernel(const float* __restrict__ pla,
                     const float* __restrict__ ppr,
                     float* __restrict__ grams) {
    const int nc     = blockIdx.x;
    const int chunk  = blockIdx.y;                 // 0..15
    const int rStart = chunk * ROWS_PER_CHUNK;
    const int rEnd   = min(rStart + ROWS_PER_CHUNK, NH);

    const int lane = threadIdx.x;                  // 0..31
    const int r    = lane & 15;                    // matrix row (A) / col (B)
    const int hi   = lane >> 4;
    const int kb   = hi << 1;                      // K base: 0 or 2

    // Loop-invariant per-lane selectors: branch-free operand construction.
    const float selLoad = (r < 9)  ? 1.f : 0.f;
    const float selOne  = (r == 9) ? 1.f : 0.f;
    const int   dy      = r / 3, dx = r - dy * 3;
    const int   nbOff   = (r < 9) ? (dy * PW + dx) : 0;   // safe addr for r>=9
    const float tailVal = (hi == 0) ? 1.f : 0.f;          // j=168 slot validity

    const float* la = pla + (size_t)nc * POOL_ELEMS;
    const float* pr = ppr + (size_t)nc * POOL_ELEMS;

    v8f gLL = {}; v8f gPP = {}; v8f gLP = {};

    for (int i = rStart; i < rEnd; ++i) {
        const float* laRow = la + i * PW + nbOff;
        const float* prRow = pr + i * PW + nbOff;

        // 42 guard-free K=4 steps covering j = 0..167
        for (int j0 = 0; j0 < NW - 1; j0 += 4) {
            v2f aL, aP;
            #pragma unroll
            for (int jj = 0; jj < 2; ++jj) {
                float lv = laRow[j0 + kb + jj];
                float pv = prRow[j0 + kb + jj];
                aL[jj] = lv * selLoad + selOne;    // rows 9: 1, rows 10-15: 0
                aP[jj] = pv * selLoad + selOne;
            }
            gLL = __builtin_amdgcn_wmma_f32_16x16x4_f32(false, aL, false, aL,
                                                        (short)0, gLL, false, false);
            gPP = __builtin_amdgcn_wmma_f32_16x16x4_f32(false, aP, false, aP,
                                                        (short)0, gPP, false, false);
            gLP = __builtin_amdgcn_wmma_f32_16x16x4_f32(false, aL, false, aP,
                                                        (short)0, gLP, false, false);
        }

        // Row tail: only sample j=168 (slot kb==0, jj==0) is valid.
        {
            float lv = laRow[NW - 1];
            float pv = prRow[NW - 1];
            v2f aL = {}, aP = {};
            aL[0] = (lv * selLoad + selOne) * tailVal;
            aP[0] = (pv * selLoad + selOne) * tailVal;
            gLL = __builtin_amdgcn_wmma_f32_16x16x4_f32(false, aL, false, aL,
                                                        (short)0, gLL, false, false);
            gPP = __builtin_amdgcn_wmma_f32_16x16x4_f32(false, aP, false, aP,
                                                        (short)0, gPP, false, false);
            gLP = __builtin_amdgcn_wmma_f32_16x16x4_f32(false, aL, false, aP,
                                                        (short)0, gLP, false, false);
        }
    }

    // C/D layout: VGPR v, lanes 0-15 -> M=v, lanes 16-31 -> M=v+8; N = lane&15
    int col = lane & 15;
    if (col < 10) {
        float* g = grams + (size_t)nc * 300;
        #pragma unroll
        for (int v = 0; v < 8; ++v) {
            int row = v + 8 * hi;
            if (row < 10) {
                atomicAdd(&g[      row * 10 + col], gLL[v]);
                atomicAdd(&g[100 + row * 10 + col], gPP[v]);
                atomicAdd(&g[200 + row * 10 + col], gLP[v]);
            }
        }
    }
}

// ------------------- per-(n,c) 9x9 inverse + Cholesky logdet + final reduce
__global__ __launch_bounds__(128)
void RMI_solve_kernel(const float* __restrict__ grams, float* __restrict__ out) {
    __shared__ float red[128];
    int t = threadIdx.x;
    float rmi = 0.f;
    if (t < NC) {
        const float* g   = grams + (size_t)t * 300;
        const float* GLL = g, * GPP = g + 100, * GLP = g + 200;
        float Mc   = GLL[9 * 10 + 9];
        float invM = 1.f / Mc;
        float SL[9], SP[9];
        #pragma unroll
        for (int d = 0; d < 9; ++d) { SL[d] = GLL[90 + d]; SP[d] = GPP[90 + d]; }

        float laC[9][9], prC[9][9], lp[9][9];
        for (int d = 0; d < 9; ++d)
            for (int e = 0; e < 9; ++e) {
                laC[d][e] = GLL[d * 10 + e] - SL[d] * SL[e] * invM;
                prC[d][e] = GPP[d * 10 + e] - SP[d] * SP[e] * invM;
                lp[d][e]  = GLP[d * 10 + e] - SL[d] * SP[e] * invM;
            }
        for (int d = 0; d < 9; ++d) prC[d][d] += 1e-3f;

        // Gauss-Jordan inverse (SPD after +alpha*I, no pivoting required)
        float inv[9][9];
        for (int d = 0; d < 9; ++d)
            for (int e = 0; e < 9; ++e) inv[d][e] = (d == e) ? 1.f : 0.f;
        for (int k = 0; k < 9; ++k) {
            float p = 1.f / prC[k][k];
            for (int e = 0; e < 9; ++e) { prC[k][e] *= p; inv[k][e] *= p; }
            for (int rr = 0; rr < 9; ++rr) if (rr != k) {
                float f = prC[rr][k];
                for (int e = 0; e < 9; ++e) {
                    prC[rr][e] -= f * prC[k][e];
                    inv[rr][e] -= f * inv[k][e];
                }
            }
        }
        // appro = laC - lp * inv * lp^T + alpha*I
        float B1[9][9];
        for (int d = 0; d < 9; ++d)
            for (int f = 0; f < 9; ++f) {
                float s = 0.f;
                for (int e = 0; e < 9; ++e) s += lp[d][e] * inv[e][f];
                B1[d][f] = s;
            }
        float A[9][9];
        for (int d = 0; d < 9; ++d)
            for (int gg = 0; gg < 9; ++gg) {
                float s = 0.f;
                for (int f = 0; f < 9; ++f) s += B1[d][f] * lp[gg][f];
                A[d][gg] = laC[d][gg] - s + ((d == gg) ? 1e-3f : 0.f);
            }
        // Cholesky + logdet ; rmi = 0.5 * 2 * sum(log(diag+1e-8))
        float L[9][9];
        for (int i = 0; i < 9; ++i)
            for (int j = 0; j <= i; ++j) {
                float s = A[i][j];
                for (int k = 0; k < j; ++k) s -= L[i][k] * L[j][k];
                if (i == j) L[i][j] = sqrtf(fmaxf(s, 0.f));
                else        L[i][j] = s / L[j][j];
            }
        float ld = 0.f;
        for (int i = 0; i < 9; ++i) ld += __logf(L[i][i] + 1e-8f);
        rmi = ld;
    }
    red[t] = rmi;
    __syncthreads();
    if (t == 0) {
        float s = 0.f;
        for (int i = 0; i < 128; ++i) s += red[i];
        // sum_c( mean_n(rmi)/9 ) = sum_all / (NBATCH * 9)
        out[0] = s / 36.f;
    }
}

// ---------------------------------------------------------------------------
extern "C" void kernel_launch(void* const* d_in, const int* in_sizes, int n_in,
                              void* d_out, int out_size, void* d_ws, size_t ws_size,
                              hipStream_t stream) {
    const float* score  = (const float*)d_in[0];  // cls_score    (4,26,512,512)
    const float* target = (const float*)d_in[1];  // new_targets  (4,26,512,512)
    const float* mask   = (const float*)d_in[2];  // valid_indices(4,512,512)

    float* ws    = (float*)d_ws;
    float* pla   = ws;                                   // 104*29241 floats
    float* ppr   = pla + (size_t)NC * POOL_ELEMS;        // 104*29241 floats
    float* grams = ppr + (size_t)NC * POOL_ELEMS;        // 104*300 floats

    int gramN = NC * 300;
    RMI_zero_kernel<<<(gramN + 255) / 256, 256, 0, stream>>>(grams, gramN);

    int poolN = NC * POOL_ELEMS;
    RMI_pool_kernel<<<(poolN + 255) / 256, 256, 0, stream>>>(score, target, mask,
                                                             pla, ppr);

    dim3 g2(NC, CHUNKS);   // 104 x 16 blocks, one wave each (uniform chunks)
    RMI_gram_kernel<<<g2, 32, 0, stream>>>(pla, ppr, grams);

    RMI_solve_kernel<<<1, 128, 0, stream>>>(grams, (float*)d_out);
}